// OhemCELoss_79027398246840
// MI455X (gfx1250) — compile-verified
//
#include <hip/hip_runtime.h>
#include <stdint.h>

// Problem constants (from reference): B=16, C=5, H=W=768
#define BATCH 16
#define NCLS  5
#define HWSZ  (768 * 768)            // 589824 pixels per image
#define NPIX  (BATCH * HWSZ)         // 9437184 total pixels

#define NBLOCKS 3072                 // streaming grid (fixed -> deterministic partials)
#define TPB     256                  // 8 waves of 32

typedef float v2f __attribute__((ext_vector_type(2)));
typedef float v8f __attribute__((ext_vector_type(8)));

// ---- workspace layout (uint32 words) ----
#define WS_NPOS    0
#define WS_NNEG    1
#define WS_NTAKE   2
#define WS_BIN1    3
#define WS_ABOVE1  4
#define WS_REM     5
#define WS_THRESH  6
#define WS_TIES    7
#define WS_HIST1   16
#define HIST1_BINS 32768
#define WS_HIST2   (WS_HIST1 + HIST1_BINS)
#define HIST2_BINS 65536
#define WS_BLKLP   (WS_HIST2 + HIST2_BINS)   // per-block loss_pos partials (float)
#define WS_BLKLN   (WS_BLKLP + NBLOCKS)      // per-block loss_neg partials (float)
#define WS_WORDS   (WS_BLKLN + NBLOCKS)

// ---------------------------------------------------------------------------
// Exact 32-lane f32 sum using V_WMMA_F32_16X16X4_F32 with ones operands.
// Stage 1: D1 = A(v per lane in slot K0/K2, zeros elsewhere) x ones(4x16)
//          -> D1[m][n] = rowsum(m) = v(m) + v(m+16), replicated over n.
// Stage 2: four accumulating WMMAs with A = ones(16x4) and B built from the
//          8 D1 accumulator VGPRs (two per call). Summing over K consumes
//          every (VGPR, lane-half) slot exactly once, so each D2 element is
//          the exact total of all 32 lane values (robust to K-slot ordering).
// ---------------------------------------------------------------------------
__device__ __forceinline__ float wave_sum_wmma(float v) {
  v2f a;   a[0] = v;    a[1] = 0.0f;
  v2f one; one[0] = 1.0f; one[1] = 1.0f;
  v8f d1 = {};
  d1 = __builtin_amdgcn_wmma_f32_16x16x4_f32(false, a, false, one, (short)0, d1, false, false);
  v8f d2 = {};
  v2f b;
  b[0] = d1[0]; b[1] = d1[1];
  d2 = __builtin_amdgcn_wmma_f32_16x16x4_f32(false, one, false, b, (short)0, d2, false, false);
  b[0] = d1[2]; b[1] = d1[3];
  d2 = __builtin_amdgcn_wmma_f32_16x16x4_f32(false, one, false, b, (short)0, d2, false, false);
  b[0] = d1[4]; b[1] = d1[5];
  d2 = __builtin_amdgcn_wmma_f32_16x16x4_f32(false, one, false, b, (short)0, d2, false, false);
  b[0] = d1[6]; b[1] = d1[7];
  d2 = __builtin_amdgcn_wmma_f32_16x16x4_f32(false, one, false, b, (short)0, d2, false, false);
  return d2[0];
}

// Deterministic block reduction: WMMA wave sums -> 8 shared slots -> serial add.
__device__ __forceinline__ float block_sum(float v, float* s_w) {
  float ws = wave_sum_wmma(v);
  unsigned wid = threadIdx.x >> 5;
  if ((threadIdx.x & 31) == 0) s_w[wid] = ws;
  __syncthreads();
  float tot = 0.0f;
  if (threadIdx.x == 0) {
    for (int i = 0; i < TPB / 32; ++i) tot += s_w[i];
  }
  __syncthreads();
  return tot;  // valid in thread 0
}

__device__ __forceinline__ float compute_ce(const float* __restrict__ pred, int i, int t) {
  int b   = i / HWSZ;
  int pix = i - b * HWSZ;
  const float* base = pred + (size_t)b * (NCLS * HWSZ) + pix;
  float p0 = base[0 * HWSZ];
  float p1 = base[1 * HWSZ];
  float p2 = base[2 * HWSZ];
  float p3 = base[3 * HWSZ];
  float p4 = base[4 * HWSZ];
  float m  = fmaxf(fmaxf(fmaxf(p0, p1), fmaxf(p2, p3)), p4);
  float s  = expf(p0 - m) + expf(p1 - m) + expf(p2 - m) + expf(p3 - m) + expf(p4 - m);
  float pt = (t == 0) ? p0 : (t == 1) ? p1 : (t == 2) ? p2 : (t == 3) ? p3 : p4;
  return logf(s) + (m - pt);   // -log softmax at target; >= 0
}

__device__ __forceinline__ unsigned ce_pattern(float ce) {
  return __float_as_uint(ce) & 0x7FFFFFFFu;   // fold -0 into +0; monotone for ce>=0
}

// ---- K0: zero workspace -----------------------------------------------------
__global__ void k_init(unsigned* __restrict__ ws) {
  int i = blockIdx.x * blockDim.x + threadIdx.x;
  if (i < WS_WORDS) ws[i] = 0u;
}

// ---- K1: CE stats: pos loss partials, pos/neg counts, coarse hist (top16) ---
__global__ void k_stats(const float* __restrict__ pred, const int* __restrict__ gt,
                        unsigned* __restrict__ ws) {
  __shared__ float s_w[TPB / 32];
  __shared__ unsigned s_cp, s_cn;
  if (threadIdx.x == 0) { s_cp = 0u; s_cn = 0u; }
  __syncthreads();

  unsigned* hist1 = ws + WS_HIST1;
  float lp = 0.0f;
  unsigned cpos = 0, cneg = 0;
  const int stride = gridDim.x * blockDim.x;
  for (int i = blockIdx.x * blockDim.x + threadIdx.x; i < NPIX; i += stride) {
    int nx = i + stride;
    if (nx < NPIX) {                       // software prefetch of next tile
      __builtin_prefetch(&gt[nx], 0, 0);   // -> global_prefetch_b8
      __builtin_prefetch(&pred[(size_t)(nx / HWSZ) * (NCLS * HWSZ) + (nx % HWSZ)], 0, 0);
    }
    int t = gt[i];
    float ce = compute_ce(pred, i, t);
    bool pos = (t >= 1) && (t <= 3);
    if (pos) { lp += ce; cpos++; }
    else {
      cneg++;
      atomicAdd(&hist1[ce_pattern(ce) >> 16], 1u);
    }
  }
  atomicAdd(&s_cp, cpos);
  atomicAdd(&s_cn, cneg);
  float blp = block_sum(lp, s_w);          // WMMA-based, exact f32
  if (threadIdx.x == 0) {
    ((float*)ws)[WS_BLKLP + blockIdx.x] = blp;
    atomicAdd(&ws[WS_NPOS], s_cp);
    atomicAdd(&ws[WS_NNEG], s_cn);
  }
}

// ---- K2: select coarse threshold bin from hist1 -----------------------------
__global__ void k_select1(const int* __restrict__ ratio, unsigned* __restrict__ ws) {
  __shared__ unsigned csum[TPB];
  const unsigned* hist1 = ws + WS_HIST1;
  const int per = HIST1_BINS / TPB;        // 128 bins per thread
  unsigned t = threadIdx.x, s = 0;
  for (int j = 0; j < per; ++j) s += hist1[t * per + j];
  csum[t] = s;
  __syncthreads();
  if (t == 0) {
    unsigned npos = ws[WS_NPOS];
    unsigned nneg = ws[WS_NNEG];
    float r = (float)ratio[0];
    unsigned cap = (unsigned)(r * (float)npos);        // trunc toward zero (matches ref)
    unsigned ntake = nneg < cap ? nneg : cap;
    ws[WS_NTAKE] = ntake;
    unsigned cum = 0;                                  // count strictly above current
    int chunk = 0;
    for (int c = TPB - 1; c >= 0; --c) {
      if (cum + csum[c] >= ntake) { chunk = c; break; }
      cum += csum[c];
    }
    int base = chunk * per, bin1 = base;
    for (int b = base + per - 1; b >= base; --b) {
      unsigned h = hist1[b];
      if (cum + h >= ntake) { bin1 = b; break; }
      cum += h;
    }
    ws[WS_BIN1]   = (unsigned)bin1;
    ws[WS_ABOVE1] = cum;
    ws[WS_REM]    = ntake - cum;           // in [0, hist1[bin1]]
  }
}

// ---- K3: fine histogram of low16 bits for values in the coarse bin ----------
__global__ void k_hist2(const float* __restrict__ pred, const int* __restrict__ gt,
                        unsigned* __restrict__ ws) {
  unsigned bin1 = ws[WS_BIN1];
  unsigned* hist2 = ws + WS_HIST2;
  const int stride = gridDim.x * blockDim.x;
  for (int i = blockIdx.x * blockDim.x + threadIdx.x; i < NPIX; i += stride) {
    int t = gt[i];
    if (!((t >= 1) && (t <= 3))) {
      unsigned pat = ce_pattern(compute_ce(pred, i, t));
      if ((pat >> 16) == bin1) atomicAdd(&hist2[pat & 0xFFFFu], 1u);
    }
  }
}

// ---- K4: exact threshold pattern + tie count --------------------------------
__global__ void k_select2(unsigned* __restrict__ ws) {
  __shared__ unsigned csum[TPB];
  const unsigned* hist2 = ws + WS_HIST2;
  const int per = HIST2_BINS / TPB;        // 256 bins per thread
  unsigned t = threadIdx.x, s = 0;
  for (int j = 0; j < per; ++j) s += hist2[t * per + j];
  csum[t] = s;
  __syncthreads();
  if (t == 0) {
    unsigned rem    = ws[WS_REM];
    unsigned bin1   = ws[WS_BIN1];
    unsigned cum = 0;
    int chunk = TPB - 1;
    for (int c = TPB - 1; c >= 0; --c) {
      if (cum + csum[c] >= rem) { chunk = c; break; }
      cum += csum[c];
    }
    int base = chunk * per, L = base + per - 1;
    for (int b = base + per - 1; b >= base; --b) {
      unsigned h = hist2[b];
      if (cum + h >= rem) { L = b; break; }
      cum += h;
    }
    ws[WS_THRESH] = (bin1 << 16) | (unsigned)L;   // exact pattern P
    ws[WS_TIES]   = rem - cum;                    // #values == P to include
  }
}

// ---- K5: sum negative CE strictly above threshold ---------------------------
__global__ void k_sumneg(const float* __restrict__ pred, const int* __restrict__ gt,
                         unsigned* __restrict__ ws) {
  __shared__ float s_w[TPB / 32];
  unsigned P = ws[WS_THRESH];
  float acc = 0.0f;
  const int stride = gridDim.x * blockDim.x;
  for (int i = blockIdx.x * blockDim.x + threadIdx.x; i < NPIX; i += stride) {
    int nx = i + stride;
    if (nx < NPIX) __builtin_prefetch(&gt[nx], 0, 0);
    int t = gt[i];
    if (!((t >= 1) && (t <= 3))) {
      float ce = compute_ce(pred, i, t);
      if (ce_pattern(ce) > P) acc += ce;
    }
  }
  float bsum = block_sum(acc, s_w);        // WMMA-based, exact f32
  if (threadIdx.x == 0) ((float*)ws)[WS_BLKLN + blockIdx.x] = bsum;
}

// ---- K6: deterministic final reduction + tie term + normalization -----------
__global__ void k_final(unsigned* __restrict__ ws, float* __restrict__ out) {
  __shared__ float s_w[TPB / 32];
  const float* blp = (const float*)ws + WS_BLKLP;
  const float* bln = (const float*)ws + WS_BLKLN;
  float lp = 0.0f, ln = 0.0f;
  for (int i = threadIdx.x; i < NBLOCKS; i += TPB) { lp += blp[i]; ln += bln[i]; }
  float tlp = block_sum(lp, s_w);
  float tln = block_sum(ln, s_w);
  if (threadIdx.x == 0) {
    unsigned ties  = ws[WS_TIES];
    unsigned P     = ws[WS_THRESH];
    unsigned npos  = ws[WS_NPOS];
    unsigned ntake = ws[WS_NTAKE];
    if (ties > 0) tln += (float)ties * __uint_as_float(P);
    out[0] = (tlp + tln) / (float)(npos + ntake);
  }
}

extern "C" void kernel_launch(void* const* d_in, const int* in_sizes, int n_in,
                              void* d_out, int out_size, void* d_ws, size_t ws_size,
                              hipStream_t stream) {
  (void)in_sizes; (void)n_in; (void)out_size; (void)ws_size;
  const float* pred  = (const float*)d_in[0];
  const int*   gt    = (const int*)d_in[1];
  const int*   ratio = (const int*)d_in[2];
  unsigned*    ws    = (unsigned*)d_ws;
  float*       out   = (float*)d_out;

  k_init   <<<(WS_WORDS + TPB - 1) / TPB, TPB, 0, stream>>>(ws);
  k_stats  <<<NBLOCKS, TPB, 0, stream>>>(pred, gt, ws);
  k_select1<<<1, TPB, 0, stream>>>(ratio, ws);
  k_hist2  <<<NBLOCKS, TPB, 0, stream>>>(pred, gt, ws);
  k_select2<<<1, TPB, 0, stream>>>(ws);
  k_sumneg <<<NBLOCKS, TPB, 0, stream>>>(pred, gt, ws);
  k_final  <<<1, TPB, 0, stream>>>(ws, out);
}